// GlobalMaxAvgPool_64733747085921
// MI455X (gfx1250) — compile-verified
//
#include <hip/hip_runtime.h>

// ---------------------------------------------------------------------------
// Segment max + mean pooling (sorted segment ids), N=2e6 x C=128 -> B x 2C.
// HBM-bound: 1.024 GB read once => ~44us floor at 23.3 TB/s. Strategy:
//  - wave-per-row-run register accumulation (sorted ids => rare flushes)
//  - non-temporal global_load_b128 streaming (stream >> 192MB L2)
//  - gfx1250 async global->LDS DMA for staging the ids (ASYNCcnt path)
//  - global float atomics only at segment boundaries / range ends
// ---------------------------------------------------------------------------

typedef float v4f __attribute__((ext_vector_type(4)));
typedef int   v4i __attribute__((ext_vector_type(4)));

constexpr int C           = 128;   // channels (reference fixed)
constexpr int TWO_C       = 256;
constexpr int THREADS     = 256;   // 8 wave32 per block
constexpr int WAVES       = THREADS / 32;
constexpr int BLOCK_ROWS  = 2048;  // rows staged per block (8KB of ids in LDS)
constexpr int ROWS_PER_WAVE = BLOCK_ROWS / WAVES; // 256

#if __has_builtin(__builtin_amdgcn_global_load_async_to_lds_b32)
#define USE_ASYNC_LDS 1
#endif

__device__ __forceinline__ void waitAsync0() {
#if __has_builtin(__builtin_amdgcn_s_wait_asynccnt)
  __builtin_amdgcn_s_wait_asynccnt(0);
#else
  asm volatile("s_wait_asynccnt 0x0" ::: "memory");
#endif
}

__device__ __forceinline__ v4f splat4(float x) {
  v4f r = {x, x, x, x};
  return r;
}

__device__ __forceinline__ v4f vmax4(v4f a, v4f b) {
  v4f r;
#pragma unroll
  for (int k = 0; k < 4; ++k) r[k] = fmaxf(a[k], b[k]);  // v_max_num_f32
  return r;
}

// Float atomic-max via signed-int max (>=0) / unsigned-int min (<0).
// Accumulator initialized to bits 0xFF800000 (-inf), valid under both orders.
__device__ __forceinline__ void atomicMaxF(float* a, float v) {
  if (v >= 0.0f) atomicMax((int*)a, __float_as_int(v));
  else           atomicMin((unsigned int*)a, __float_as_uint(v));
}

__device__ __forceinline__ void flushSeg(int seg, int lane, const v4f mx, const v4f sm,
                                         float cnt, float* __restrict__ maxacc,
                                         float* __restrict__ sumacc,
                                         float* __restrict__ counts) {
  if (seg < 0 || cnt <= 0.0f) return;
  float* mp = maxacc + (size_t)seg * C + lane * 4;
  float* sp = sumacc + (size_t)seg * C + lane * 4;
#pragma unroll
  for (int k = 0; k < 4; ++k) {
    atomicMaxF(mp + k, mx[k]);
    atomicAdd(sp + k, sm[k]);
  }
  if (lane == 0) atomicAdd(counts + seg, cnt);
}

__global__ __launch_bounds__(THREADS) void seg_pool_init(unsigned int* __restrict__ maxaccBits,
                                                         float* __restrict__ sumacc,
                                                         float* __restrict__ counts,
                                                         int BC, int B) {
  const int i = blockIdx.x * blockDim.x + threadIdx.x;
  if (i < BC) {
    maxaccBits[i] = 0xFF800000u;  // -inf
    sumacc[i]     = 0.0f;
  }
  if (i < B) counts[i] = 0.0f;
}

__global__ __launch_bounds__(THREADS) void seg_pool_main(const float* __restrict__ feat,
                                                         const int* __restrict__ ids,
                                                         float* __restrict__ maxacc,
                                                         float* __restrict__ sumacc,
                                                         float* __restrict__ counts,
                                                         int N) {
  __shared__ __align__(16) int sIds[BLOCK_ROWS];

  const int tid       = threadIdx.x;
  const int blockBase = blockIdx.x * BLOCK_ROWS;
  const int blockEnd  = min(N, blockBase + BLOCK_ROWS);

  // ---- Stage this block's batch_ids into LDS (gfx1250 async DMA path) ----
  for (int i = tid; i < BLOCK_ROWS; i += THREADS) {
    const int g = blockBase + i;
    if (g < N) {
#ifdef USE_ASYNC_LDS
      void* gp = (void*)(ids + g);
      void* lp = (void*)(&sIds[i]);
      __builtin_amdgcn_global_load_async_to_lds_b32(
          (__attribute__((address_space(1))) int*)gp,
          (__attribute__((address_space(3))) int*)lp,
          /*offset=*/0, /*cpol=*/0);
#else
      sIds[i] = ids[g];
#endif
    }
  }
#ifdef USE_ASYNC_LDS
  waitAsync0();
#endif
  __syncthreads();

  // ---- Register-resident run accumulation over this wave's row range ----
  const int wave = tid >> 5;
  const int lane = tid & 31;
  int r          = blockBase + wave * ROWS_PER_WAVE;
  const int rEnd = min(blockEnd, r + ROWS_PER_WAVE);

  const float NEG_INF = -__builtin_inff();
  v4f  mx  = splat4(NEG_INF);
  v4f  sm  = splat4(0.0f);
  float cnt = 0.0f;
  int  cur = -1;

  const float* colBase = feat + (size_t)lane * 4;  // this lane's 4 channels

  // 4-row unrolled main loop: issue 4 independent NT b128 loads, then fold.
  for (; r + 3 < rEnd; r += 4) {
    const v4i s4 = *(const v4i*)&sIds[r - blockBase];  // ds_load_b128 (aligned)
    v4f vv[4];
    vv[0] = __builtin_nontemporal_load((const v4f*)(colBase + (size_t)(r + 0) * C));
    vv[1] = __builtin_nontemporal_load((const v4f*)(colBase + (size_t)(r + 1) * C));
    vv[2] = __builtin_nontemporal_load((const v4f*)(colBase + (size_t)(r + 2) * C));
    vv[3] = __builtin_nontemporal_load((const v4f*)(colBase + (size_t)(r + 3) * C));
#pragma unroll
    for (int k = 0; k < 4; ++k) {
      const int seg = s4[k];
      if (seg != cur) {  // wave-uniform, rare (ids sorted)
        flushSeg(cur, lane, mx, sm, cnt, maxacc, sumacc, counts);
        cur = seg;
        mx  = splat4(NEG_INF);
        sm  = splat4(0.0f);
        cnt = 0.0f;
      }
      mx = vmax4(mx, vv[k]);
      sm += vv[k];
      cnt += 1.0f;
    }
  }
  // Remainder rows (tail block only).
  for (; r < rEnd; ++r) {
    const int seg = sIds[r - blockBase];
    const v4f v = __builtin_nontemporal_load((const v4f*)(colBase + (size_t)r * C));
    if (seg != cur) {
      flushSeg(cur, lane, mx, sm, cnt, maxacc, sumacc, counts);
      cur = seg;
      mx  = splat4(NEG_INF);
      sm  = splat4(0.0f);
      cnt = 0.0f;
    }
    mx = vmax4(mx, v);
    sm += v;
    cnt += 1.0f;
  }
  flushSeg(cur, lane, mx, sm, cnt, maxacc, sumacc, counts);
}

__global__ __launch_bounds__(THREADS) void seg_pool_finalize(const float* __restrict__ maxacc,
                                                             const float* __restrict__ sumacc,
                                                             const float* __restrict__ counts,
                                                             float* __restrict__ out,
                                                             int total) {
  const int i = blockIdx.x * blockDim.x + threadIdx.x;
  if (i >= total) return;
  const int b = i / TWO_C;
  const int j = i - b * TWO_C;
  if (j < C) {
    out[i] = maxacc[b * C + j];
  } else {
    out[i] = sumacc[b * C + (j - C)] / fmaxf(counts[b], 1.0f);
  }
}

extern "C" void kernel_launch(void* const* d_in, const int* in_sizes, int n_in,
                              void* d_out, int out_size, void* d_ws, size_t ws_size,
                              hipStream_t stream) {
  const float* feat = (const float*)d_in[0];
  const int*   ids  = (const int*)d_in[1];
  const int N  = in_sizes[1];          // number of rows
  const int B  = out_size / TWO_C;     // segments, from output shape [B, 2C]
  const int BC = B * C;

  float* maxacc = (float*)d_ws;        // [B, C]
  float* sumacc = maxacc + BC;         // [B, C]
  float* counts = sumacc + BC;         // [B]

  seg_pool_init<<<(BC + THREADS - 1) / THREADS, THREADS, 0, stream>>>(
      (unsigned int*)maxacc, sumacc, counts, BC, B);

  const int grid = (N + BLOCK_ROWS - 1) / BLOCK_ROWS;
  seg_pool_main<<<grid, THREADS, 0, stream>>>(feat, ids, maxacc, sumacc, counts, N);

  seg_pool_finalize<<<(out_size + THREADS - 1) / THREADS, THREADS, 0, stream>>>(
      maxacc, sumacc, counts, (float*)d_out, out_size);
}